// GCNLayer_75557064671959
// MI455X (gfx1250) — compile-verified
//
#include <hip/hip_runtime.h>
#include <hip/hip_bf16.h>
#include <stdint.h>

// out[row_e] += val_e * x[col_e]   (COO SpMM, fp32, D=128)
// Memory-bound: ~135 MB HBM (x/out live in 192MB L2) => WMMA inapplicable.
// CDNA5 paths: global_load_async_to_lds_b32 (ASYNCcnt, double-buffered edge
// streaming), s_wait_asynccnt, ds_load_b128 broadcast, global_load_b128
// gathers, global_atomic_add_f32 scatter-adds.

#define D_FEAT 128
#define TILE 32          // edges per wave-tile (one edge slot per lane)
#define WAVES_PER_BLOCK 8

__global__ __launch_bounds__(256) void zero_out_kernel(float4* __restrict__ out4, int n4) {
    int i = blockIdx.x * blockDim.x + threadIdx.x;
    const int stride = gridDim.x * blockDim.x;
    const float4 z = make_float4(0.f, 0.f, 0.f, 0.f);
    for (; i < n4; i += stride) out4[i] = z;
}

__global__ __launch_bounds__(256) void spmm_scatter_kernel(
    const float4* __restrict__ x4,     // [N, 32] float4 view of x [N,128]
    const int*    __restrict__ e_row,  // [E]
    const int*    __restrict__ e_col,  // [E]
    const float*  __restrict__ e_val,  // [E]
    float*        __restrict__ out,    // [N, 128]
    int nEdges, int nTiles)
{
    // 16 B per edge slot: {row, col, val, pad} -> ds_load_b128 friendly.
    __shared__ uint32_t edg[WAVES_PER_BLOCK][2][TILE][4];

    const int lane       = threadIdx.x & 31;
    const int wv         = threadIdx.x >> 5;
    const int waveId     = blockIdx.x * WAVES_PER_BLOCK + wv;
    const int waveStride = gridDim.x * WAVES_PER_BLOCK;

    const uint64_t rowBase = (uint64_t)(uintptr_t)e_row;
    const uint64_t colBase = (uint64_t)(uintptr_t)e_col;
    const uint64_t valBase = (uint64_t)(uintptr_t)e_val;

    // Stage tile t's 32 edges into LDS buffer b via the async-to-LDS engine.
    // 3 async b32 instructions per call -> ASYNCcnt += 3 for this wave.
    auto stage = [&](int t, int b) {
        int e = t * TILE + lane;
        uint32_t ec  = (e < nEdges) ? (uint32_t)e : 0u;   // clamp: always legal
        uint32_t off = ec * 4u;                            // byte offset
        uint32_t lds = (uint32_t)(uintptr_t)(&edg[wv][b][lane][0]); // LDS byte addr
        asm volatile("global_load_async_to_lds_b32 %0, %1, %2"
                     :: "v"(lds),      "v"(off), "s"(rowBase) : "memory");
        asm volatile("global_load_async_to_lds_b32 %0, %1, %2"
                     :: "v"(lds + 4u), "v"(off), "s"(colBase) : "memory");
        asm volatile("global_load_async_to_lds_b32 %0, %1, %2"
                     :: "v"(lds + 8u), "v"(off), "s"(valBase) : "memory");
    };

    // Process one staged edge slot: broadcast read, gather, scale, scatter-add.
    auto process = [&](int b, int s) {
        uint4 ew = *(const uint4*)&edg[wv][b][s][0];   // uniform-addr ds_load_b128
        const int   row = (int)ew.x;
        const int   col = (int)ew.y;
        const float val = __uint_as_float(ew.z);

        float4 xv = x4[(size_t)col * (D_FEAT / 4) + lane];  // global_load_b128
        float4 m;
        m.x = val * xv.x; m.y = val * xv.y;
        m.z = val * xv.z; m.w = val * xv.w;

        float* o = out + (size_t)row * D_FEAT + (size_t)lane * 4;
        unsafeAtomicAdd(o + 0, m.x);   // global_atomic_add_f32 (no return)
        unsafeAtomicAdd(o + 1, m.y);
        unsafeAtomicAdd(o + 2, m.z);
        unsafeAtomicAdd(o + 3, m.w);
    };

    int t = waveId;
    if (t < nTiles) stage(t, 0);      // preamble: 3 outstanding
    int cb = 0;

    for (; t < nTiles; t += waveStride, cb ^= 1) {
        // Always stage a "next" buffer (clamped) so the ASYNCcnt invariant
        // (6 outstanding before the wait) holds on every iteration.
        int nt = t + waveStride;
        stage(nt < nTiles ? nt : t, cb ^ 1);
        __builtin_amdgcn_s_wait_asynccnt(3);   // oldest 3 (current buf) done

        const int base = t * TILE;
        const int rem  = nEdges - base;
        if (rem >= TILE) {
            // Fast path: fixed trip count, unrolled x4 so the compiler can
            // batch 4 independent ds_load_b128 + global_load_b128 gathers
            // ahead of the atomic groups (more LOADcnt in flight per wave).
            #pragma unroll 4
            for (int s = 0; s < TILE; ++s) process(cb, s);
        } else {
            for (int s = 0; s < rem; ++s) process(cb, s);
        }
    }
    __builtin_amdgcn_s_wait_asynccnt(0);  // drain dummy stages before endpgm
}

extern "C" void kernel_launch(void* const* d_in, const int* in_sizes, int n_in,
                              void* d_out, int out_size, void* d_ws, size_t ws_size,
                              hipStream_t stream) {
    const float* x  = (const float*)d_in[0];           // [N,128]
    const int*   ei = (const int*)d_in[1];             // [2,E] row-major
    const float* ev = (const float*)d_in[2];           // [E]
    float*       out = (float*)d_out;

    const int E = in_sizes[2];
    const int nTiles = (E + TILE - 1) / TILE;

    // 1) zero the (poisoned) output
    int n4 = out_size / 4;
    int zblocks = (n4 + 255) / 256;
    if (zblocks > 4096) zblocks = 4096;
    if (zblocks < 1) zblocks = 1;
    zero_out_kernel<<<zblocks, 256, 0, stream>>>((float4*)out, n4);

    // 2) scatter-add SpMM. 2048 blocks x 8 waves = 16384 waves over 50000 tiles.
    int blocks = 2048;
    if (blocks > nTiles) blocks = (nTiles + WAVES_PER_BLOCK - 1) / WAVES_PER_BLOCK;
    if (blocks < 1) blocks = 1;
    spmm_scatter_kernel<<<blocks, 256, 0, stream>>>(
        (const float4*)x, ei, ei + E, ev, out, E, nTiles);
}